// DemographicParityGap_4913442587183
// MI455X (gfx1250) — compile-verified
//
#include <hip/hip_runtime.h>

// DemographicParityGap on MI455X (gfx1250, wave32).
//
// class_sums[c,s] = sum_{n: bp[n]==c} output[n,s]  ==  onehot(bp)^T @ output
// implemented with V_WMMA_F32_16X16X4_F32: 4 rows of `output` per WMMA,
// exact f32 products (one-hot entries exact), f32 accumulation.
//
// Stage 1: W waves, each iterates over disjoint 16-row chunks. Every lane
//          loads exactly the two bp values its A-slot needs (broadcast b32,
//          constant offsets after unroll -> no addressable array, no LDS
//          detour, no selects) plus 8 predicated NT b32 loads of `output`;
//          then 4 chained WMMAs. One EXEC toggle + one load wait per chunk.
// Stage 2: 80 blocks tree-reduce one cell each over W partials (deterministic).
// Stage 3: 1 thread: row tail (N%16) + demP + 28 pairwise gaps -> -loss.

typedef __attribute__((ext_vector_type(2))) float v2f;
typedef __attribute__((ext_vector_type(8))) float v8f;

#define CLS 10
#define SUB 8
#define CELLS (CLS * SUB)   // 80

__global__ void dpg_stage1(const float* __restrict__ outp,
                           const int* __restrict__ bp,
                           float* __restrict__ partials,  // [CELLS][W]
                           int nChunks16, int W)
{
    const int lane = threadIdx.x & 31;
    const int wave = (int)((blockIdx.x * blockDim.x + threadIdx.x) >> 5);
    const int m    = lane & 15;        // A: class row  | B/D: column (subgroup)
    const int half = lane >> 4;        // 0: K slots 0/1 ; 1: K slots 2/3
    const bool colOK = (m < SUB);      // only 8 of 16 B columns are real
    const int k0 = 2 * half;           // K offset held in VGPR0 of A/B
    const int k1 = 2 * half + 1;       // K offset held in VGPR1 of A/B

    // Per-lane base pointers: lane directly loads the bp value its A-slot
    // needs (broadcast within each half-wave -> single cacheline fetch).
    const int* __restrict__ bpk0 = bp + k0;
    const int* __restrict__ bpk1 = bp + k1;

    v8f acc = {};                      // D/C tile: 16 classes x 16 cols

    for (int c16 = wave; c16 < nChunks16; c16 += W) {
        const long base = (long)c16 * 16;          // 16 rows per iteration

        // ---- issue ALL loads for this chunk first (deep pipeline) ----
        int a0i[4], a1i[4];
#pragma unroll
        for (int j = 0; j < 4; ++j) {              // constant offsets -> SROA
            a0i[j] = bpk0[base + 4 * j];
            a1i[j] = bpk1[base + 4 * j];
        }

        float bx[4], by[4];
#pragma unroll
        for (int j = 0; j < 4; ++j) { bx[j] = 0.0f; by[j] = 0.0f; }
        if (colOK) {                               // single EXEC toggle / chunk
#pragma unroll
            for (int j = 0; j < 4; ++j) {
                const long r = base + 4 * j;
                bx[j] = __builtin_nontemporal_load(outp + (r + k0) * SUB + m);
                by[j] = __builtin_nontemporal_load(outp + (r + k1) * SUB + m);
            }
        }

        // ---- 4 chained WMMAs (K=4 slabs) ----
#pragma unroll
        for (int j = 0; j < 4; ++j) {
            // A (16x4 one-hot): VGPR0 holds K=k0, VGPR1 holds K=k1
            v2f a;
            a.x = (a0i[j] == m) ? 1.0f : 0.0f;
            a.y = (a1i[j] == m) ? 1.0f : 0.0f;
            v2f b; b.x = bx[j]; b.y = by[j];
            // D = A x B + C  (neg_a, A, neg_b, B, c_mod, C, reuse_a, reuse_b)
            acc = __builtin_amdgcn_wmma_f32_16x16x4_f32(
                      false, a, false, b, (short)0, acc, false, false);
        }
    }

    // D layout: VGPR r, this lane -> (class = r + 8*half, subgroup = m)
    if (colOK) {
#pragma unroll
        for (int r = 0; r < 8; ++r) {
            const int c = r + 8 * half;
            if (c < CLS)
                partials[(long)(c * SUB + m) * W + wave] = acc[r];
        }
    }
}

__global__ void dpg_stage2(const float* __restrict__ partials,
                           float* __restrict__ cs, int W)
{
    __shared__ float red[256];
    const int cell = blockIdx.x;                  // 0..79
    const float* p = partials + (long)cell * W;
    float s = 0.0f;
    for (int i = threadIdx.x; i < W; i += 256) s += p[i];
    red[threadIdx.x] = s;
    __syncthreads();
    for (int off = 128; off > 0; off >>= 1) {
        if ((int)threadIdx.x < off) red[threadIdx.x] += red[threadIdx.x + off];
        __syncthreads();
    }
    if (threadIdx.x == 0) cs[cell] = red[0];
}

__global__ void dpg_stage3(const float* __restrict__ outp,
                           const int* __restrict__ bp,
                           const float* __restrict__ cs_in,
                           float* __restrict__ res,
                           int n, int tailStart)
{
    if (threadIdx.x != 0 || blockIdx.x != 0) return;

    float cs[CLS][SUB];
#pragma unroll
    for (int c = 0; c < CLS; ++c)
#pragma unroll
        for (int s = 0; s < SUB; ++s) cs[c][s] = cs_in[c * SUB + s];

    // rows not covered by the 16-row WMMA chunks (N % 16, usually 0)
    for (int r = tailStart; r < n; ++r) {
        const int c = bp[r];
#pragma unroll
        for (int s = 0; s < SUB; ++s) cs[c][s] += outp[(long)r * SUB + s];
    }

    float col[SUB];
#pragma unroll
    for (int s = 0; s < SUB; ++s) {
        float t = 0.0f;
        for (int c = 0; c < CLS; ++c) t += cs[c][s];   // col_sums = sum over classes
        col[s] = t;
    }

    float loss = 0.0f;
    for (int c = 0; c < CLS; ++c) {
        float d[SUB];
#pragma unroll
        for (int s = 0; s < SUB; ++s) d[s] = cs[c][s] / col[s];
#pragma unroll
        for (int i = 0; i < SUB; ++i)
#pragma unroll
            for (int j2 = i + 1; j2 < SUB; ++j2) {
                const float g = d[i] - d[j2];
                loss += g * g;
            }
    }
    const int ncombs = SUB * (SUB - 1) / 2;       // 28
    res[0] = -(loss / (float)(CLS * ncombs));
}

extern "C" void kernel_launch(void* const* d_in, const int* in_sizes, int n_in,
                              void* d_out, int out_size, void* d_ws, size_t ws_size,
                              hipStream_t stream)
{
    const float* outp = (const float*)d_in[0];   // [N, 8] f32
    const int*   bp   = (const int*)d_in[1];     // [N] int32
    // d_in[2] labels: unused by the reference loss
    // d_in[3]/d_in[4]: num_classes=10 / num_subgroups=8 (compile-time constants here)

    const int n = in_sizes[1];
    const int nChunks16 = n / 16;
    const int tailStart = nChunks16 * 16;

    // Pick wave count W (multiple of 8 = waves/block), shrink if d_ws is small.
    int W = 8192;
    {
        long avail = (long)(ws_size / 4) - (CELLS + 16);
        if ((long)CELLS * W > avail) {
            W = (int)(avail / CELLS);
            W &= ~7;
            if (W < 8) W = 8;
        }
    }

    float* partials = (float*)d_ws;                    // [80][W]
    float* cs       = partials + (long)CELLS * W;      // [80]

    const int blocks = W / 8;                          // 256 thr = 8 wave32/block
    dpg_stage1<<<blocks, 256, 0, stream>>>(outp, bp, partials, nChunks16, W);
    dpg_stage2<<<CELLS, 256, 0, stream>>>(partials, cs, W);
    dpg_stage3<<<1, 32, 0, stream>>>(outp, bp, cs, (float*)d_out, n, tailStart);
}